// _Entangle_layer_41369124995659
// MI455X (gfx1250) — compile-verified
//
#include <hip/hip_runtime.h>
#include <cstdint>

// Gate constants:
//   c  = 2^-0.25          (X-gate scale, real)
//   zr = 2^-0.75 = Re(z) = Im(z),  z = i^0.5 * 2^-0.25   (Z-gate phase)
#define C0 0.8408964152537145f
#define ZR 0.5946035575013605f

__device__ __forceinline__ float4 scl(const float4 v, float s) {
  return make_float4(v.x * s, v.y * s, v.z * s, v.w * s);
}
__device__ __forceinline__ float4 yxwz(const float4 v) {   // i ^ 1 within a float4
  return make_float4(v.y, v.x, v.w, v.z);
}
__device__ __forceinline__ float4 addf4(const float4 a, const float4 b) {
  return make_float4(a.x + b.x, a.y + b.y, a.z + b.z, a.w + b.w);
}
__device__ __forceinline__ float4 subf4(const float4 a, const float4 b) {
  return make_float4(a.x - b.x, a.y - b.y, a.z - b.z, a.w - b.w);
}
__device__ __forceinline__ float4 altsign(const float4 v) { // (+,-,+,-) for bit0 sign
  return make_float4(v.x, -v.y, v.z, -v.w);
}

// Input : re/im  [2048, 8, 8, 64]  -> 131072 rows of 64 floats (16 float4)
// Output: [2 (re/im), 2048, 8, 16, 64]
// Thread t: g4 = t&7 (owns float4 groups g4 and g4+8 of its row), row = t>>3.
__global__ __launch_bounds__(256, 1) void entangle_kernel(
    const float4* __restrict__ re_in,
    const float4* __restrict__ im_in,
    float4* __restrict__ out4)
{
  __shared__ float4 smem[1024];            // 16 KB: 4 float4 per thread

  const unsigned tid = threadIdx.x;
  const unsigned t   = blockIdx.x * 256u + tid;
  const unsigned g4  = t & 7u;             // float4 group in lower half of row
  const unsigned row = t >> 3;             // (b*8 + blk)*8 + d
  const unsigned d   = row & 7u;
  const unsigned blk = (row >> 3) & 7u;
  const unsigned b   = row >> 6;

  const unsigned ibase = row * 16u + g4;   // float4 index of lo group; hi = +8

  // ---- CDNA5 async copy path: global -> LDS, tracked by ASYNCcnt ----
  const uint32_t l0 = (uint32_t)(uintptr_t)(&smem[tid * 4u]);
  const unsigned long long a_re_lo = (unsigned long long)(uintptr_t)(re_in + ibase);
  const unsigned long long a_re_hi = (unsigned long long)(uintptr_t)(re_in + ibase + 8u);
  const unsigned long long a_im_lo = (unsigned long long)(uintptr_t)(im_in + ibase);
  const unsigned long long a_im_hi = (unsigned long long)(uintptr_t)(im_in + ibase + 8u);
  asm volatile("global_load_async_to_lds_b128 %0, %1, off" :: "v"(l0),        "v"(a_re_lo) : "memory");
  asm volatile("global_load_async_to_lds_b128 %0, %1, off" :: "v"(l0 + 16u), "v"(a_re_hi) : "memory");
  asm volatile("global_load_async_to_lds_b128 %0, %1, off" :: "v"(l0 + 32u), "v"(a_im_lo) : "memory");
  asm volatile("global_load_async_to_lds_b128 %0, %1, off" :: "v"(l0 + 48u), "v"(a_im_hi) : "memory");
  asm volatile("s_wait_asynccnt 0" ::: "memory");

  const float4 re_lo = smem[tid * 4u + 0u];
  const float4 re_hi = smem[tid * 4u + 1u];
  const float4 im_lo = smem[tid * 4u + 2u];
  const float4 im_hi = smem[tid * 4u + 3u];

  float4 tre_lo, tre_hi, tim_lo, tim_hi;   // top half: Ux @ in  (scaled permute)
  float4 bre_lo, bre_hi, bim_lo, bim_hi;   // bot half: Uz @ in  (signed phase)
  if ((blk & 1u) == 0u) {
    // qubit 0: kron(U, I32) -> swap = i^32 (group pair), sign from bit5
    tre_lo = scl(re_hi, C0);  tre_hi = scl(re_lo, C0);
    tim_lo = scl(im_hi, C0);  tim_hi = scl(im_lo, C0);
    bre_lo = scl(subf4(re_lo, im_lo),  ZR);
    bim_lo = scl(addf4(re_lo, im_lo),  ZR);
    bre_hi = scl(subf4(re_hi, im_hi), -ZR);
    bim_hi = scl(addf4(re_hi, im_hi), -ZR);
  } else {
    // qubit 5: kron(I32, U) -> swap = i^1 (in-register shuffle), sign from bit0
    tre_lo = scl(yxwz(re_lo), C0);  tre_hi = scl(yxwz(re_hi), C0);
    tim_lo = scl(yxwz(im_lo), C0);  tim_hi = scl(yxwz(im_hi), C0);
    bre_lo = altsign(scl(subf4(re_lo, im_lo), ZR));
    bim_lo = altsign(scl(addf4(re_lo, im_lo), ZR));
    bre_hi = altsign(scl(subf4(re_hi, im_hi), ZR));
    bim_hi = altsign(scl(addf4(re_hi, im_hi), ZR));
  }

  // Output row base (float4 units): [b, blk, dd, i] with dd=d (top), d+8 (bot).
  const unsigned orow  = b * 2048u + blk * 256u + d * 16u;
  const unsigned OUT4C = 4194304u;         // re/im component stride in float4

  out4[orow + g4]                     = tre_lo;
  out4[orow + g4 + 8u]                = tre_hi;
  out4[orow + 128u + g4]              = bre_lo;
  out4[orow + 128u + g4 + 8u]         = bre_hi;
  out4[OUT4C + orow + g4]             = tim_lo;
  out4[OUT4C + orow + g4 + 8u]        = tim_hi;
  out4[OUT4C + orow + 128u + g4]      = bim_lo;
  out4[OUT4C + orow + 128u + g4 + 8u] = bim_hi;
}

extern "C" void kernel_launch(void* const* d_in, const int* in_sizes, int n_in,
                              void* d_out, int out_size, void* d_ws, size_t ws_size,
                              hipStream_t stream) {
  (void)in_sizes; (void)n_in; (void)out_size; (void)d_ws; (void)ws_size;
  const float4* re = (const float4*)d_in[0];
  const float4* im = (const float4*)d_in[1];
  float4* out = (float4*)d_out;
  // 131072 rows * 8 group-pairs = 1,048,576 threads = 4096 blocks x 256
  hipLaunchKernelGGL(entangle_kernel, dim3(4096), dim3(256), 0, stream, re, im, out);
}